// FCOSMultiStrideFilter_15719580303963
// MI455X (gfx1250) — compile-verified
//
#include <hip/hip_runtime.h>
#include <cstdint>

// FCOS multi-stride filter for MI455X (gfx1250).
// Pure bandwidth-bound NCHW->NHWC transpose + channel-max mask + masked write.
// Uses CDNA5 async global->LDS loads (ASYNCcnt path) to perform the transpose
// inside the data mover: coalesced global reads, strided LDS scatter.

#define TILE     128   // spatial locations per block
#define NTHREADS 128   // 4 wave32 per block (one full WGP pass)
#define ROW      87    // 2 coord + 80 cls + 4 bbox + 1 ctr
#define NCLS     80

// Per-lane async copy of 4 bytes: global (VGPR pair addr) -> LDS (VGPR byte addr).
// CDNA5 ISA: GLOBAL_LOAD_ASYNC_TO_LDS_B32, tracked with ASYNCcnt.
__device__ __forceinline__ void async_load_b32(unsigned lds_byte_addr, const void* gptr) {
    asm volatile("global_load_async_to_lds_b32 %0, %1, off"
                 :: "v"(lds_byte_addr),
                    "v"((unsigned long long)(uintptr_t)gptr)
                 : "memory");
}

__device__ __forceinline__ void wait_async_zero() {
    asm volatile("s_wait_asynccnt 0" ::: "memory");
}

__global__ __launch_bounds__(NTHREADS)
void fcos_filter_level(const float* __restrict__ cls,
                       const float* __restrict__ bb,
                       const float* __restrict__ ctr,
                       float* __restrict__ out,
                       int HW, int W, int stride_px)
{
    // LDS staging buffer in the *output* layout: tile[loc][ROW].
    // ROW=87 is odd -> stride coprime with 64 banks -> conflict-free.
    __shared__ float tile[TILE * ROW];
    __shared__ float maskv[TILE];

    const int t         = threadIdx.x;
    const int tile_base = blockIdx.x * TILE;
    const int b         = blockIdx.y;
    const int nvalid    = min(TILE, HW - tile_base);

    // ---- Phase 1: async transpose-copy global -> LDS ------------------------
    // Lane t owns output row t. For each channel c, the 32 lanes of a wave read
    // 32 consecutive floats of plane c (coalesced) and the async engine scatters
    // them to LDS at stride ROW*4 bytes. Channel slot in the row is c+2.
    if (t < nvalid) {
        const unsigned lds_row = (unsigned)(uintptr_t)(&tile[t * ROW]);
        const size_t   gidx    = (size_t)tile_base + (size_t)t;
        #pragma unroll
        for (int c = 0; c < NCLS; ++c)
            async_load_b32(lds_row + 4u * (unsigned)(2 + c),
                           cls + ((size_t)b * NCLS + c) * (size_t)HW + gidx);
        #pragma unroll
        for (int c = 0; c < 4; ++c)
            async_load_b32(lds_row + 4u * (unsigned)(82 + c),
                           bb + ((size_t)b * 4 + c) * (size_t)HW + gidx);
        async_load_b32(lds_row + 4u * 86u,
                       ctr + (size_t)b * (size_t)HW + gidx);
    }
    wait_async_zero();     // this wave's async LDS writes are complete
    __syncthreads();       // make them visible across the workgroup

    // ---- Phase 2: per-location max over 80 class scores, mask, coords -------
    if (t < nvalid) {
        const float* row = &tile[t * ROW];
        float m = row[2];
        #pragma unroll
        for (int c = 1; c < NCLS; ++c)
            m = fmaxf(m, row[2 + c]);
        maskv[t] = (m > 0.5f) ? 1.0f : 0.0f;

        const int gidx = tile_base + t;
        const int y    = gidx / W;
        const int x    = gidx - y * W;
        const int half = stride_px >> 1;
        tile[t * ROW + 0] = (float)(x * stride_px + half);
        tile[t * ROW + 1] = (float)(y * stride_px + half);
    }
    __syncthreads();

    // ---- Phase 3: coalesced masked store ------------------------------------
    // Block's output region is contiguous: [tile_base, tile_base+nvalid) x ROW.
    float* __restrict__ dst = out + ((size_t)b * (size_t)HW + (size_t)tile_base) * ROW;
    const int total = nvalid * ROW;
    for (int j = t; j < total; j += NTHREADS) {
        dst[j] = tile[j] * maskv[j / ROW];
    }
}

extern "C" void kernel_launch(void* const* d_in, const int* in_sizes, int n_in,
                              void* d_out, int out_size, void* d_ws, size_t ws_size,
                              hipStream_t stream) {
    (void)in_sizes; (void)n_in; (void)out_size; (void)d_ws; (void)ws_size;

    static const int HS[5]      = {160, 80, 40, 20, 10};
    static const int STRIDES[5] = {8, 16, 32, 64, 128};
    const int B = 8;

    float* out = (float*)d_out;
    size_t off = 0;
    for (int l = 0; l < 5; ++l) {
        const float* cls  = (const float*)d_in[3 * l + 0];
        const float* bbox = (const float*)d_in[3 * l + 1];
        const float* ctr  = (const float*)d_in[3 * l + 2];
        const int W  = HS[l];
        const int HW = W * W;
        dim3 grid((unsigned)((HW + TILE - 1) / TILE), (unsigned)B);
        fcos_filter_level<<<grid, NTHREADS, 0, stream>>>(
            cls, bbox, ctr, out + off, HW, W, STRIDES[l]);
        off += (size_t)B * (size_t)HW * (size_t)ROW;
    }
}